// Sample_11501922419073
// MI455X (gfx1250) — compile-verified
//
#include <hip/hip_runtime.h>
#include <stdint.h>

// im2col for 8x(512x512x1) -> 8x(505x505x64), VALID 8x8 window, stride 1.
// out[b,h,w,i] = in[b, h+i/8, w+i%8, 0]   (one-hot conv kernel == identity)
//
// Memory-bound: 522 MB stores vs 8 MB (L2-resident) input. Strategy:
//   TDM (tensor_load_to_lds) stages an 11x71 f32 input tile into LDS,
//   then threads gather from LDS and emit fully-coalesced float4 stores.

#define IN_H   512
#define IN_W   512
#define OUT_H  505
#define OUT_W  505
#define WIN    8
#define WT     64                  // output w-positions per block
#define HT     4                   // output h-rows per block
#define TILE_W (WT + WIN - 1)      // 71 input cols staged
#define TILE_H (HT + WIN - 1)      // 11 input rows staged
#define NQ     (HT * WT * 16)      // float4 quads per block tile = 4096

typedef __attribute__((ext_vector_type(4))) unsigned int u32x4;
typedef __attribute__((ext_vector_type(8))) int          i32x8;
typedef __attribute__((ext_vector_type(4))) int          i32x4;

__global__ __launch_bounds__(256)
void im2col_tdm_kernel(const float* __restrict__ in, float4* __restrict__ out)
{
    __shared__ float tile[TILE_H * TILE_W];

    const int b      = blockIdx.z;
    const int wstart = blockIdx.x * WT;
    const int hstart = blockIdx.y * HT;
    // Clamp staged tile so it always lies fully inside the 512x512 tensor
    // (never rely on TDM OOB behavior).
    const int w0 = (wstart > IN_W - TILE_W) ? (IN_W - TILE_W) : wstart;
    const int h0 = (hstart > IN_H - TILE_H) ? (IN_H - TILE_H) : hstart;
    const int tid = threadIdx.x;

    const uint64_t gbase =
        (uint64_t)(uintptr_t)(in + ((size_t)b * IN_H + h0) * IN_W + w0);

#if __has_builtin(__builtin_amdgcn_tensor_load_to_lds)
    // ---- Tensor Data Mover: async 2D tile load Global -> LDS ----
    // Low 32 bits of a generic pointer to LDS are the LDS byte offset
    // (SHARED aperture lives entirely in the high word).
    const uint32_t ldsb = (uint32_t)(uintptr_t)(void*)&tile[0];

    // D# Group 0 (128b): count=1 | lds_addr | global_addr[56:0] | type=2
    u32x4 g0;
    g0[0] = 1u;
    g0[1] = ldsb;
    g0[2] = (uint32_t)gbase;
    g0[3] = ((uint32_t)(gbase >> 32) & 0x01FFFFFFu) | (2u << 30);

    // D# Group 1 (256b): wg_mask=0, data_size=2 (4B), no barrier/iterate/pad,
    // tensor_dim0=512, tensor_dim1=512, tile_dim0=71, tile_dim1=11,
    // tile_dim2=0, tensor_dim0_stride=512, tensor_dim1_stride=0 (2D).
    i32x8 g1;
    g1[0] = 0x00020000;                                        // data_size=4B
    g1[1] = (int)((uint32_t)IN_W << 16);                       // tdim0[15:0]
    g1[2] = (int)(((uint32_t)IN_W >> 16) |
                  ((uint32_t)IN_H << 16));                     // tdim0 hi|tdim1 lo
    g1[3] = (int)(((uint32_t)IN_H >> 16) |
                  ((uint32_t)TILE_W << 16));                   // tdim1 hi|tile0
    g1[4] = (int)TILE_H;                                       // tile1 | tile2=0
    g1[5] = (int)IN_W;                                         // dim0_stride lo
    g1[6] = 0;                                                 // stride0 hi|stride1 lo
    g1[7] = 0;                                                 // stride1 hi
    i32x4 gz = {0, 0, 0, 0};                                   // groups 2/3 unused (2D)

    if (tid < 32) {                                            // one wave issues DMA
#if __clang_major__ >= 23
        i32x8 gz8 = {0, 0, 0, 0, 0, 0, 0, 0};
        __builtin_amdgcn_tensor_load_to_lds(g0, g1, gz, gz, gz8, 0);
#else
        __builtin_amdgcn_tensor_load_to_lds(g0, g1, gz, gz, 0);
#endif
        __builtin_amdgcn_s_wait_tensorcnt(0);
    }
    __syncthreads();
#else
    // ---- Fallback: cooperative staged load ----
    for (int i = tid; i < TILE_H * TILE_W; i += 256) {
        int r = i / TILE_W, c = i - r * TILE_W;
        tile[i] = in[((size_t)b * IN_H + (h0 + r)) * IN_W + (w0 + c)];
    }
    __syncthreads();
#endif

    // ---- Gather from LDS, stream coalesced float4 stores ----
    // q = (hl, wl, c4) with c4 fastest: consecutive threads -> consecutive
    // 16B output slots -> each wave32 writes 512 contiguous bytes.
    for (int q = tid; q < NQ; q += 256) {
        const int c4 = q & 15;           // channel quad: i = 4*c4..4*c4+3
        const int wl = (q >> 4) & (WT - 1);
        const int hl = q >> 10;
        const int w = wstart + wl;
        const int h = hstart + hl;
        if (w >= OUT_W || h >= OUT_H) continue;

        const int dy  = c4 >> 1;         // window row   (i/8)
        const int dx0 = (c4 & 1) << 2;   // window col base (i%8 in {0,4})
        const float* s = &tile[(h - h0 + dy) * TILE_W + (w - w0) + dx0];
        float4 v;
        v.x = s[0]; v.y = s[1]; v.z = s[2]; v.w = s[3];

        out[(((size_t)b * OUT_H + h) * OUT_W + w) * 16 + c4] = v;
    }
}

extern "C" void kernel_launch(void* const* d_in, const int* in_sizes, int n_in,
                              void* d_out, int out_size, void* d_ws, size_t ws_size,
                              hipStream_t stream) {
    const float* in  = (const float*)d_in[0];   // (8,512,512,1) f32
    // d_in[1] is the one-hot HWIO kernel == im2col identity; not needed.
    float4* out = (float4*)d_out;               // (8,505,505,64) f32

    dim3 grid((OUT_W + WT - 1) / WT,            // 8 w-tiles
              (OUT_H + HT - 1) / HT,            // 127 h-tiles
              8);                               // batch
    im2col_tdm_kernel<<<grid, 256, 0, stream>>>(in, out);
}